// DMM_28063316313018
// MI455X (gfx1250) — compile-verified
//
#include <hip/hip_runtime.h>
#include <math.h>

// ELBO for guide-RNN DMM.  T=1000 sequential, B=16384 parallel (one lane per
// batch element => 512 wave32s).  Memory-bound: 327.7 MB @ 23.3 TB/s ~ 14 us.
// CDNA5 paths used:
//  * V_WMMA_F32_16X16X4_F32 for the x @ W_ih^T input projection (A = W_ih
//    padded to 16x4, B = 16 batch columns; ds_swizzle SWAPX16 marshals the
//    K=2 row / result routing between wave halves).
//  * GLOBAL_LOAD_ASYNC_TO_LDS_B128 triple-buffered chunk pipeline (8
//    timesteps/chunk, 40 KB/buffer) gated by s_wait_asynccnt, to keep ~5 MB
//    in flight at the forced 1-wave/SIMD occupancy.
//  * v_tanh_f32 via __builtin_amdgcn_tanhf (trans op) instead of branchy OCML.

#define T_STEPS 1000
#define BATCH   16384
#define SIGMA_F 0.01f

#define BLK     256
#define CHUNK   8                          // timesteps per LDS chunk
#define NCHUNK  (T_STEPS / CHUNK)          // 125
#define DROW    (BLK * 12)                 // data bytes per t-row per block (3072)
#define EROW    (BLK * 8)                  // eps  bytes per t-row per block (2048)
#define DBYTES  (CHUNK * DROW)             // 24576
#define EBYTES  (CHUNK * EROW)             // 16384
#define BUFB    (DBYTES + EBYTES)          // 40960 per buffer
#define DITER   (DBYTES / 16 / BLK)        // 6  B128 ops/thread (data)
#define EITER   (EBYTES / 16 / BLK)        // 4  B128 ops/thread (eps)
#define NI      (DITER + EITER)            // 10 async ops/thread/chunk

typedef __attribute__((ext_vector_type(2))) float v2f;
typedef __attribute__((ext_vector_type(8))) float v8f;

__device__ __forceinline__ float swap_half(float v) {
    return __int_as_float(__builtin_amdgcn_ds_swizzle(__float_as_int(v), 0x401F));
}

__device__ __forceinline__ float fast_tanh(float x) {
#if __has_builtin(__builtin_amdgcn_tanhf)
    return __builtin_amdgcn_tanhf(x);      // v_tanh_f32
#else
    return tanhf(x);
#endif
}

#if __has_builtin(__builtin_amdgcn_s_wait_asynccnt)
#define WAIT_ASYNC(n) __builtin_amdgcn_s_wait_asynccnt(n)
#else
#define WAIT_ASYNC(n) asm volatile("s_wait_asynccnt %0" ::"i"(n) : "memory")
#endif

// Async copy of 16 bytes from the global address gptr to the LDS offset lds_off; bumps ASYNCcnt
__device__ __forceinline__ void async_b128(unsigned lds_off, const void* gptr) {
    asm volatile("global_load_async_to_lds_b128 %0, %1, off"
                 :: "v"(lds_off), "v"(gptr)
                 : "memory");
}

// u[m] = sum_k W_ih[m][k] * x[k] per lane, via V_WMMA_F32_16X16X4_F32.
__device__ __forceinline__ void proj_wmma(v2f Aop, bool hi,
                                          float x0, float x1, float x2,
                                          float& u0, float& u1) {
    float sx0 = swap_half(x0);
    float sx1 = swap_half(x1);
    float sx2 = swap_half(x2);
    v2f B1, B2;
    B1.x = hi ? sx2 : x0;   // batches 0..15:  VGPR0 = K0 row | K2 row
    B1.y = hi ? 0.f : x1;   //                 VGPR1 = K1 row | K3 pad
    B2.x = hi ? x2  : sx0;  // batches 16..31
    B2.y = hi ? 0.f : sx1;
    v8f cz = {};
    v8f D1 = __builtin_amdgcn_wmma_f32_16x16x4_f32(
        false, Aop, false, B1, (short)0, cz, false, false);
    v8f D2 = __builtin_amdgcn_wmma_f32_16x16x4_f32(
        false, Aop, false, B2, (short)0, cz, false, false);
    float s0 = swap_half(D2[0]);
    float s1 = swap_half(D2[1]);
    u0 = hi ? s0 : D1[0];
    u1 = hi ? s1 : D1[1];
}

__global__ __launch_bounds__(BLK) void DMM_elbo_main(
    const float* __restrict__ data,  // [T,B,3]
    const float* __restrict__ eps,   // [T,B,2]
    const float* __restrict__ W_ih,
    const float* __restrict__ W_hh,
    const float* __restrict__ b_ih,
    const float* __restrict__ b_hh,
    const float* __restrict__ h0,
    const float* __restrict__ z0,
    const float* __restrict__ Wt,
    const float* __restrict__ btv,
    const float* __restrict__ We,
    const float* __restrict__ be,
    double* __restrict__ partial)
{
    __shared__ char   lds_buf[3 * BUFB];   // 120 KB triple-buffered stream
    __shared__ double sred[BLK];

    const int tid  = threadIdx.x;
    const int b0   = blockIdx.x * BLK;
    const int lane = tid & 31;
    const bool hi  = lane >= 16;
    const int l16  = lane & 15;

    // Uniform weights
    const float w00 = W_ih[0], w01 = W_ih[1], w02 = W_ih[2];
    const float w10 = W_ih[3], w11 = W_ih[4], w12 = W_ih[5];
    const float a00 = W_hh[0], a01 = W_hh[1], a10 = W_hh[2], a11 = W_hh[3];
    const float bias0 = b_ih[0] + b_hh[0];
    const float bias1 = b_ih[1] + b_hh[1];
    const float t00 = Wt[0], t01 = Wt[1], t10 = Wt[2], t11 = Wt[3];
    const float bt0 = btv[0], bt1 = btv[1];
    const float e00 = We[0], e01 = We[1], e10 = We[2];
    const float e11 = We[3], e20 = We[4], e21 = We[5];
    const float be0 = be[0], be1 = be[1], be2 = be[2];

    // WMMA A operand: rows M=0,1 hold W_ih.  A 16x4 f32 layout:
    // VGPR0 = K0 (lanes0-15) / K2 (lanes16-31); VGPR1 = K1 / K3.
    v2f Aop;
    Aop.x = hi ? (l16 == 0 ? w02 : (l16 == 1 ? w12 : 0.f))
               : (l16 == 0 ? w00 : (l16 == 1 ? w10 : 0.f));
    Aop.y = hi ? 0.f
               : (l16 == 0 ? w01 : (l16 == 1 ? w11 : 0.f));

    const float K2c = 0.5f / (SIGMA_F * SIGMA_F);

    float hc0 = h0[0], hc1 = h0[1];
    float zp0 = z0[0], zp1 = z0[1];
    double acc = 0.0;

    // Byte bases for this block's 256-wide batch slab
    const size_t drow = (size_t)BATCH * 12;   // bytes per t-row (global)
    const size_t erow = (size_t)BATCH * 8;
    const char* dbase = (const char*)data + (size_t)b0 * 12;
    const char* ebase = (const char*)eps  + (size_t)b0 * 8;
    const unsigned lbase = (unsigned)(size_t)(void*)lds_buf;  // LDS byte offset

    // ---- async chunk issue: copy CHUNK t-rows of data+eps into buffer ----
    auto issue_chunk = [&](int c) {
        const char* dg = dbase + (size_t)c * CHUNK * drow;
        const char* eg = ebase + (size_t)c * CHUNK * erow;
        const unsigned lb = lbase + (unsigned)((c % 3) * BUFB);
        #pragma unroll
        for (int k = 0; k < DITER; ++k) {           // DROW/16 = 192 units/row
            int u = tid + k * BLK;
            int row = u / 192, col = u - row * 192;
            async_b128(lb + (unsigned)u * 16u,
                       dg + (size_t)row * drow + (size_t)col * 16);
        }
        #pragma unroll
        for (int k = 0; k < EITER; ++k) {           // EROW/16 = 128 units/row
            int u = tid + k * BLK;
            int row = u / 128, col = u - row * 128;
            async_b128(lb + DBYTES + (unsigned)u * 16u,
                       eg + (size_t)row * erow + (size_t)col * 16);
        }
    };

    issue_chunk(0);
    issue_chunk(1);

    for (int c = 0; c < NCHUNK; ++c) {
        if (c + 2 < NCHUNK) {
            issue_chunk(c + 2);
            WAIT_ASYNC(2 * NI);   // two newer chunks outstanding -> chunk c done
        } else if (c + 1 < NCHUNK) {
            WAIT_ASYNC(NI);
        } else {
            WAIT_ASYNC(0);
        }
        __syncthreads();          // all waves' portions of chunk c landed

        const float* ldsd = (const float*)(lds_buf + (c % 3) * BUFB);
        const float* ldse = (const float*)(lds_buf + (c % 3) * BUFB + DBYTES);

        #pragma unroll
        for (int j = 0; j < CHUNK; ++j) {
            const int t = c * CHUNK + j;
            float x0v = ldsd[(j * BLK + tid) * 3 + 0];
            float x1v = ldsd[(j * BLK + tid) * 3 + 1];
            float x2v = ldsd[(j * BLK + tid) * 3 + 2];
            float ev0 = ldse[(j * BLK + tid) * 2 + 0];
            float ev1 = ldse[(j * BLK + tid) * 2 + 1];

            if (t > 0) {  // uniform branch; t=0 only advances the RNN
                // ELBO index t-1: q_loc = hc (rnn_out[t-1]) with eps[t], data[t]
                float zv0 = fmaf(SIGMA_F, ev0, hc0);
                float zv1 = fmaf(SIGMA_F, ev1, hc1);
                float zl0 = fmaf(t00, zp0, fmaf(t01, zp1, bt0));
                float zl1 = fmaf(t10, zp0, fmaf(t11, zp1, bt1));
                float xl0 = fmaf(e00, zv0, fmaf(e01, zv1, be0));
                float xl1 = fmaf(e10, zv0, fmaf(e11, zv1, be1));
                float xl2 = fmaf(e20, zv0, fmaf(e21, zv1, be2));
                float dz0 = zv0 - zl0, dz1 = zv1 - zl1;
                float dx0 = x0v - xl0, dx1 = x1v - xl1, dx2 = x2v - xl2;
                float quad = dz0 * dz0 + dz1 * dz1 + dx0 * dx0 + dx1 * dx1 + dx2 * dx2;
                float s = fmaf(0.5f, ev0 * ev0 + ev1 * ev1, -K2c * quad);
                acc += (double)s;
                zp0 = zv0;
                zp1 = zv1;
            }

            // RNN advance with data[t] (WMMA input projection + hw tanh)
            float u0, u1;
            proj_wmma(Aop, hi, x0v, x1v, x2v, u0, u1);
            float p0 = u0 + bias0 + a00 * hc0 + a01 * hc1;
            float p1 = u1 + bias1 + a10 * hc0 + a11 * hc1;
            hc0 = fast_tanh(p0);
            hc1 = fast_tanh(p1);
        }
        __syncthreads();          // buffer fully consumed before it is refilled
    }

    // Deterministic block reduction (fixed tree order)
    sred[tid] = acc;
    __syncthreads();
    for (int s = 128; s > 0; s >>= 1) {
        if (tid < s) sred[tid] += sred[tid + s];
        __syncthreads();
    }
    if (tid == 0) partial[blockIdx.x] = sred[0];
}

__global__ void DMM_elbo_final(const double* __restrict__ partial, int n,
                               float* __restrict__ out, double constTerm) {
    if (blockIdx.x == 0 && threadIdx.x == 0) {
        double s = 0.0;
        for (int i = 0; i < n; ++i) s += partial[i];  // fixed order: deterministic
        out[0] = (float)(s + constTerm);
    }
}

extern "C" void kernel_launch(void* const* d_in, const int* in_sizes, int n_in,
                              void* d_out, int out_size, void* d_ws, size_t ws_size,
                              hipStream_t stream) {
    (void)in_sizes; (void)n_in; (void)out_size; (void)ws_size;
    const float* data = (const float*)d_in[0];
    const float* eps  = (const float*)d_in[1];
    const float* W_ih = (const float*)d_in[2];
    const float* W_hh = (const float*)d_in[3];
    const float* b_ih = (const float*)d_in[4];
    const float* b_hh = (const float*)d_in[5];
    const float* h0   = (const float*)d_in[6];
    const float* z0   = (const float*)d_in[7];
    const float* Wt   = (const float*)d_in[8];
    const float* btv  = (const float*)d_in[9];
    const float* We   = (const float*)d_in[10];
    const float* be   = (const float*)d_in[11];
    float*  out     = (float*)d_out;
    double* partial = (double*)d_ws;

    const int threads = BLK;
    const int blocks  = BATCH / BLK;  // 64

    DMM_elbo_main<<<blocks, threads, 0, stream>>>(
        data, eps, W_ih, W_hh, b_ih, b_hh, h0, z0, Wt, btv, We, be, partial);

    // Net analytic constant: per (t,b), logp contributes 5 copies of
    // (-log(sigma)-0.5*log(2*pi)), logq subtracts 2 -> net +3 per (t,b).
    const double C = -(log(0.01) + 0.5 * log(2.0 * M_PI));
    const double constTerm = (double)(T_STEPS - 1) * (double)BATCH * 3.0 * C;

    DMM_elbo_final<<<1, 64, 0, stream>>>(partial, blocks, out, constTerm);
}